// RecSysGNN_72395968742053
// MI455X (gfx1250) — compile-verified
//
#include <hip/hip_runtime.h>
#include <hip/hip_bf16.h>
#include <math.h>

#define DD 64
#define NEG_SLOPE 0.2f
#define SM_EPS 1e-16f

typedef __attribute__((ext_vector_type(2))) float v2f;
typedef __attribute__((ext_vector_type(8))) float v8f;

// float atomic max via signed/unsigned integer atomics (works across sign mix,
// init value must be -inf)
__device__ __forceinline__ void atomic_max_float(float* addr, float val) {
    if (val >= 0.0f) atomicMax((int*)addr, __float_as_int(val));
    else             atomicMin((unsigned int*)addr, __float_as_uint(val));
}

// -------------------- kernels --------------------

__global__ void k_copy_init(const float* __restrict__ emb, float* __restrict__ out1,
                            float* __restrict__ out2, int nd) {
    int i = blockIdx.x * blockDim.x + threadIdx.x;
    if (i < nd) { float v = emb[i]; out1[i] = v; out2[i] = v; }
}

__global__ void k_layer_init(float* __restrict__ next, float* __restrict__ mmax,
                             float* __restrict__ dnm, int nd, int n) {
    int i = blockIdx.x * blockDim.x + threadIdx.x;
    if (i < nd) next[i] = 0.0f;
    if (i < n) { mmax[i] = -__builtin_inff(); dnm[i] = 0.0f; }
}

// Y = X @ W, X:[N,64] f32 row-major, W:[64,64] f32 row-major.
// One wave -> 16x64 output strip using V_WMMA_F32_16X16X4_F32.
// W is staged in LDS in a K-pair-permuted layout so each B fragment is a
// single aligned float2 -> one ds_load_b64 straight into an even VGPR pair.
__global__ __launch_bounds__(256)
void k_gemm64(const float* __restrict__ X, const float* __restrict__ W,
              float* __restrict__ Y, int nstrips, int ntot) {
    __shared__ float wlds[DD * DD];     // wlds[kp*128 + c*2 + j] = W[2*kp+j][c]
    int t = threadIdx.x;
    for (int i = t; i < DD * DD; i += 256) {
        int kp  = i >> 7;               // K pair 0..31
        int rem = i & 127;
        int c   = rem >> 1;             // column 0..63
        int j   = rem & 1;              // element of pair
        wlds[i] = W[(2 * kp + j) * DD + c];
    }
    __syncthreads();

    int wave = t >> 5, lane = t & 31;
    int strip = blockIdx.x * 8 + wave;
    if (strip >= nstrips) return;       // wave-uniform exit, EXEC stays all-1s
    long rowBase = (long)strip * 16;

    int mrow = lane & 15;
    int kh   = lane >> 4;               // 0: K {0,1}, 1: K {2,3} of each 4-tile
    long rowA = rowBase + mrow;
    if (rowA >= ntot) rowA = ntot - 1;  // clamp (keeps EXEC full for WMMA)
    const float* xrow = X + rowA * DD;
    int ncol = lane & 15;

    v8f c0 = {}, c1 = {}, c2 = {}, c3 = {};
#pragma unroll
    for (int kk = 0; kk < 16; ++kk) {
        int k0 = kk * 4 + kh * 2;
        v2f a; a.x = xrow[k0]; a.y = xrow[k0 + 1];
        int kp = kk * 2 + kh;           // K pair matching this lane half
        const v2f* brow = (const v2f*)&wlds[kp * 128];
        v2f b0 = brow[ 0 + ncol];
        v2f b1 = brow[16 + ncol];
        v2f b2 = brow[32 + ncol];
        v2f b3 = brow[48 + ncol];
        c0 = __builtin_amdgcn_wmma_f32_16x16x4_f32(false, a, false, b0, (short)0, c0, false, false);
        c1 = __builtin_amdgcn_wmma_f32_16x16x4_f32(false, a, false, b1, (short)0, c1, false, false);
        c2 = __builtin_amdgcn_wmma_f32_16x16x4_f32(false, a, false, b2, (short)0, c2, false, false);
        c3 = __builtin_amdgcn_wmma_f32_16x16x4_f32(false, a, false, b3, (short)0, c3, false, false);
    }

    int rofs = (lane >> 4) << 3;        // 0 or 8 (C/D VGPR layout, lane halves)
#pragma unroll
    for (int r = 0; r < 8; ++r) {
        long row = rowBase + rofs + r;
        if (row < ntot) {
            float* yr = Y + row * DD;
            yr[ 0 + ncol] = c0[r];
            yr[16 + ncol] = c1[r];
            yr[32 + ncol] = c2[r];
            yr[48 + ncol] = c3[r];
        }
    }
}

// s[e] = leakyrelu( dot(Y[dst], X[src]) + exp(scale*ea)*64 + bias ); atomic max into m[src]
__global__ void k_score(const int* __restrict__ src, const int* __restrict__ dst,
                        const float* __restrict__ ea, const float* __restrict__ Y,
                        const float* __restrict__ X, const float* __restrict__ scale,
                        const float* __restrict__ bias, int l,
                        float* __restrict__ s, float* __restrict__ mmax, int E) {
    int t = blockIdx.x * blockDim.x + threadIdx.x;
    int e = t >> 4;                     // 16 lanes per edge: 256B coalesced rows
    if (e >= E) return;
    int lane4 = t & 15;
    int sv = src[e], dv = dst[e];
    float4 yv = ((const float4*)(Y + (long)dv * DD))[lane4];
    float4 xv = ((const float4*)(X + (long)sv * DD))[lane4];
    float p = yv.x * xv.x + yv.y * xv.y + yv.z * xv.z + yv.w * xv.w;
    p += __shfl_xor(p, 1, 32);
    p += __shfl_xor(p, 2, 32);
    p += __shfl_xor(p, 4, 32);
    p += __shfl_xor(p, 8, 32);
    if (lane4 == 0) {
        float sval = p + __expf(scale[0] * ea[e]) * (float)DD + bias[l];
        sval = sval >= 0.0f ? sval : NEG_SLOPE * sval;
        s[e] = sval;
        atomic_max_float(&mmax[sv], sval);
    }
}

// s[e] = exp(s[e]-m[src]); denom[src] += s[e]
__global__ void k_expsum(const int* __restrict__ src, float* __restrict__ s,
                         const float* __restrict__ mmax, float* __restrict__ dnm, int E) {
    int e = blockIdx.x * blockDim.x + threadIdx.x;
    if (e >= E) return;
    int sv = src[e];
    float t = __expf(s[e] - mmax[sv]);
    s[e] = t;
    unsafeAtomicAdd(&dnm[sv], t);
}

// next[dst] += (s[e]/(denom[src]+eps)) * X[src]
__global__ void k_aggregate(const int* __restrict__ src, const int* __restrict__ dst,
                            const float* __restrict__ s, const float* __restrict__ dnm,
                            const float* __restrict__ X, float* __restrict__ next, int E) {
    int t = blockIdx.x * blockDim.x + threadIdx.x;
    int e = t >> 4;
    if (e >= E) return;
    int lane4 = t & 15;
    int sv = src[e], dv = dst[e];
    float alpha = s[e] / (dnm[sv] + SM_EPS);
    float4 xv = ((const float4*)(X + (long)sv * DD))[lane4];
    float* o = next + (long)dv * DD + lane4 * 4;
    unsafeAtomicAdd(o + 0, alpha * xv.x);
    unsafeAtomicAdd(o + 1, alpha * xv.y);
    unsafeAtomicAdd(o + 2, alpha * xv.z);
    unsafeAtomicAdd(o + 3, alpha * xv.w);
}

__global__ void k_accadd(const float* __restrict__ next, float* __restrict__ out2, int nd) {
    int i = blockIdx.x * blockDim.x + threadIdx.x;
    if (i < nd) out2[i] += next[i];
}

__global__ void k_scale(float* __restrict__ out2, float f, int nd) {
    int i = blockIdx.x * blockDim.x + threadIdx.x;
    if (i < nd) out2[i] *= f;
}

// -------------------- launcher --------------------

extern "C" void kernel_launch(void* const* d_in, const int* in_sizes, int n_in,
                              void* d_out, int out_size, void* d_ws, size_t ws_size,
                              hipStream_t stream) {
    const int*   edge_index = (const int*)d_in[0];     // [2,E] int32
    const float* edge_attrs = (const float*)d_in[1];   // [E]
    const float* emb        = (const float*)d_in[2];   // [N,64]
    const float* attn_w     = (const float*)d_in[3];   // [3,64,64]
    const float* attn_b     = (const float*)d_in[4];   // [3]
    const float* scale      = (const float*)d_in[5];   // [1]

    int E  = in_sizes[0] / 2;
    int N  = in_sizes[2] / DD;
    int ND = N * DD;
    const int* src = edge_index;       // row 0
    const int* dst = edge_index + E;   // row 1

    float* out1 = (float*)d_out;       // returned emb_table
    float* out2 = out1 + ND;           // returned mean accumulation

    float* ws   = (float*)d_ws;
    float* Y    = ws;                                  // N*D
    float* embA = Y + (size_t)ND;                      // N*D
    float* embB = embA + (size_t)ND;                   // N*D
    float* sbuf = embB + (size_t)ND;                   // E
    float* mbuf = sbuf + (size_t)E;                    // N
    float* dbuf = mbuf + (size_t)N;                    // N

    dim3 blk(256);
    int gnd     = (ND + 255) / 256;
    int gE16    = (int)(((long)E * 16 + 255) / 256);
    int gE      = (E + 255) / 256;
    int nstrips = (N + 15) / 16;
    int ggemm   = (nstrips + 7) / 8;

    k_copy_init<<<gnd, blk, 0, stream>>>(emb, out1, out2, ND);

    const float* cur = emb;
    float* bufs[2] = { embA, embB };
    for (int l = 0; l < 3; ++l) {
        float* next = bufs[l & 1];
        k_layer_init<<<gnd, blk, 0, stream>>>(next, mbuf, dbuf, ND, N);
        k_gemm64<<<ggemm, blk, 0, stream>>>(cur, attn_w + (size_t)l * DD * DD, Y, nstrips, N);
        k_score<<<gE16, blk, 0, stream>>>(src, dst, edge_attrs, Y, cur, scale, attn_b, l,
                                          sbuf, mbuf, E);
        k_expsum<<<gE, blk, 0, stream>>>(src, sbuf, mbuf, dbuf, E);
        k_aggregate<<<gE16, blk, 0, stream>>>(src, dst, sbuf, dbuf, cur, next, E);
        k_accadd<<<gnd, blk, 0, stream>>>(next, out2, ND);
        cur = next;
    }
    k_scale<<<gnd, blk, 0, stream>>>(out2, 0.25f, ND);
}